// ModelNew_10522669875248
// MI455X (gfx1250) — compile-verified
//
#include <hip/hip_runtime.h>

// Graph-attention style softmax aggregation, fixed degree 16, FEAT=64.
// Memory-bound gather: node_value (12.8 MB) lives in L2 (192 MB), so we hide
// L2 gather latency with CDNA5 async global->LDS copies (ASYNCcnt pipeline),
// double-buffered per wave. Softmax + weighted accumulate in f32 VALU
// (WMMA would waste 15/16 MACs on a block-diagonal operand).
// Round 2: weight broadcast via v_readlane (SGPR scalar FMA operand) instead
// of ds_bpermute, freeing the LDS pipeline for async writes + tile reads.

#define DEG   16
#define FEATD 64
#define ROW_BYTES   (FEATD * 4)        // 256 B per value row
#define TILE_FLOATS (DEG * FEATD)      // 1024 floats = 4 KB per node tile
#define WAVES_PER_BLOCK 4
#define BLOCK_THREADS   (WAVES_PER_BLOCK * 32)
#define NUM_BLOCKS      2048           // 8192 persistent waves

__global__ __launch_bounds__(BLOCK_THREADS)
void gat_softmax_agg(const int* __restrict__ col_idx,
                     const float* __restrict__ scores,
                     const float* __restrict__ value,
                     float* __restrict__ out,
                     int n_nodes, int total_waves)
{
    // 4 waves * 2 buffers * 4 KB = 32 KB LDS per block
    __shared__ float smem[WAVES_PER_BLOCK * 2 * TILE_FLOATS];

    const int lane = threadIdx.x & 31;
    const int wib  = threadIdx.x >> 5;
    const int wid  = blockIdx.x * WAVES_PER_BLOCK + wib;

    float* bufp[2];
    bufp[0] = smem + (wib * 2 + 0) * TILE_FLOATS;
    bufp[1] = smem + (wib * 2 + 1) * TILE_FLOATS;
    // HW uses addr[31:0] of the flat address as the LDS address (ISA 10.2).
    unsigned lbase[2];
    lbase[0] = (unsigned)(size_t)bufp[0];
    lbase[1] = (unsigned)(size_t)bufp[1];

    const char* vbase = (const char*)value;

    // ---- issue 8 async B128 gathers: 16 rows x 256 B into LDS buffer b ----
    auto issue_gather = [&](int node, int b) {
        // lane L holds col index of neighbor row (L & 15); halves duplicate
        int c = col_idx[node * DEG + (lane & 15)];
        unsigned lb = lbase[b];
#pragma unroll
        for (int j = 0; j < 8; ++j) {
            int idx = j * 32 + lane;          // 0..255 16-byte chunks
            int r   = idx >> 4;               // neighbor row 0..15
            int ch  = idx & 15;               // 16B chunk within row
            int rc  = __shfl(c, r, 32);       // col[r] (r differs per half)
            const void* ga = vbase + (size_t)(unsigned)rc * ROW_BYTES + ch * 16;
            unsigned la = lb + r * ROW_BYTES + ch * 16;
            asm volatile("global_load_async_to_lds_b128 %0, %1, off"
                         :: "v"(la), "v"(ga) : "memory");
        }
    };

    // ---- softmax over 16 scores + weighted accumulate from LDS buffer ----
    auto compute = [&](int node, int b) {
        float s = scores[node * DEG + (lane & 15)];
        float m = s;
#pragma unroll
        for (int d = 1; d < 16; d <<= 1)
            m = fmaxf(m, __shfl_xor(m, d, 32));
        float ex = __expf(s - m);
        float sum = ex;
#pragma unroll
        for (int d = 1; d < 16; d <<= 1)
            sum += __shfl_xor(sum, d, 32);
        float w = ex / sum;                   // lane L: weight of edge (L&15)

        const float* bp = bufp[b];
        float ax = 0.f, ay = 0.f;
#pragma unroll
        for (int k = 0; k < DEG; ++k) {
            // constant lane index -> v_readlane_b32 into SGPR; FMA uses the
            // weight as a scalar operand (no ds_bpermute on the LDS pipe)
            float wk = __int_as_float(
                __builtin_amdgcn_readlane(__float_as_int(w), k));
            float2 v = ((const float2*)(bp + k * FEATD))[lane];
            ax = fmaf(wk, v.x, ax);
            ay = fmaf(wk, v.y, ay);
        }
        float2 o; o.x = ax; o.y = ay;
        ((float2*)(out + (size_t)node * FEATD))[lane] = o;
    };

    // ---- double-buffered pipeline over this wave's nodes ----
    int node = wid;
    int b = 0;
    if (node < n_nodes) issue_gather(node, 0);
    while (node < n_nodes) {
        int next = node + total_waves;
        if (next < n_nodes) {
            issue_gather(next, b ^ 1);
            // 8 newer gathers may remain outstanding; older tile is complete
            asm volatile("s_wait_asynccnt 8" ::: "memory");
        } else {
            asm volatile("s_wait_asynccnt 0" ::: "memory");
        }
        compute(node, b);
        node = next;
        b ^= 1;
    }
}

extern "C" void kernel_launch(void* const* d_in, const int* in_sizes, int n_in,
                              void* d_out, int out_size, void* d_ws, size_t ws_size,
                              hipStream_t stream) {
    // inputs: row_ptr[int32, N+1], col_idx[int32, E], edge_scores[f32, E],
    //         node_value[f32, N*64]
    const int*   col_idx = (const int*)d_in[1];
    const float* scores  = (const float*)d_in[2];
    const float* value   = (const float*)d_in[3];
    float*       out     = (float*)d_out;

    int n_nodes = in_sizes[0] - 1;          // fixed-degree CSR
    int total_waves = NUM_BLOCKS * WAVES_PER_BLOCK;

    gat_softmax_agg<<<NUM_BLOCKS, BLOCK_THREADS, 0, stream>>>(
        col_idx, scores, value, out, n_nodes, total_waves);
}